// PageConfidenceHead_14559939133951
// MI455X (gfx1250) — compile-verified
//
#include <hip/hip_runtime.h>
#include <hip/hip_bf16.h>

// ---------------------------------------------------------------------------
// PageConfidenceHead, fully algebraically fused for MI455X (gfx1250).
// B=4096, N=128, P=64, H=128, NH=4, HD=32.
//
// Since the attention query is a single learned vector (batch-invariant) and
// the K/V path is affine in state_pages/blend, we precompute:
//   A   (64x16):  scaled score weights per head (cols 0..3 used, rest zero)
//   cS,k0 (4):    scaled blend/bias score terms
//   Gt  (256x32): W1 @ W_out @ (W_v @ W_page), stored k-major for coalescing
//   U   (32x4):   W1 @ W_out @ (W_v @ W_blend) per head
//   h1c (32):     constant bias path through V/out-proj/MLP
// Main kernel is then bandwidth-bound: 134 MB read once -> ~5.8 us roofline.
// Scores use V_WMMA_F32_16X16X4_F32 (exact f32 math, 16 WMMA / wave).
// ---------------------------------------------------------------------------

typedef __attribute__((ext_vector_type(2))) float v2f;
typedef __attribute__((ext_vector_type(8))) float v8f;

#define Hdim   128
#define Pdim   64
#define Ndim   128
#define SCALE  0.17677669529663687f  // 1/sqrt(HD=32)

// workspace layout (float offsets)
#define WS_A     0       // 64x16, B-matrix layout A[p*16 + j]
#define WS_CS    1024    // 4
#define WS_K0    1028    // 4
#define WS_H1C   1032    // 32
#define WS_UPB   1064    // 32x4  U[o*4+h]
#define WS_GT    1192    // 256x32 Gt[k*32+o]
#define WS_TOTAL 9384

// ---------------------------------------------------------------------------
// Kernel 1: fold all weights. One block, 256 threads. ~2M MACs, negligible.
// ---------------------------------------------------------------------------
__global__ __launch_bounds__(256) void pch_prep_kernel(
    const float* __restrict__ query,   // (1,H)
    const float* __restrict__ W_page,  // (H,P)
    const float* __restrict__ b_page,  // (H)
    const float* __restrict__ W_blend, // (H,1)
    const float* __restrict__ b_blend, // (H)
    const float* __restrict__ ipw,     // (3H,H)
    const float* __restrict__ ipb,     // (3H)
    const float* __restrict__ opw,     // (H,H)
    const float* __restrict__ opb,     // (H)
    const float* __restrict__ W1,      // (32,H)
    const float* __restrict__ b1,      // (32)
    float* __restrict__ ws)
{
    __shared__ float q[Hdim];
    __shared__ float tkh[4 * Hdim];     // q_h^T Wk  per head
    __shared__ float bsum[Hdim];        // b_page + b_blend
    __shared__ float bveff[Hdim];
    __shared__ float attc[Hdim];
    __shared__ float wvb[Hdim];         // Wv @ W_blend
    __shared__ float Y[32 * Hdim];      // W1 @ Wout
    __shared__ float Wvp[Hdim * Pdim];  // Wv @ W_page

    const int t = threadIdx.x;

    if (t < Hdim) {
        bsum[t] = b_page[t] + b_blend[t];
        float acc = ipb[t];                       // q projection (Wq rows 0..H-1)
        for (int j = 0; j < Hdim; ++j) acc += query[j] * ipw[t * Hdim + j];
        q[t] = acc;
    }
    __syncthreads();

    // tkh[h][j] = sum_d q[h*32+d] * Wk[h*32+d][j]
    for (int e = t; e < 4 * Hdim; e += 256) {
        int h = e >> 7, j = e & 127;
        float acc = 0.f;
        for (int d = 0; d < 32; ++d)
            acc += q[h * 32 + d] * ipw[(Hdim + h * 32 + d) * Hdim + j];
        tkh[e] = acc;
    }
    __syncthreads();

    { // A[p][h] (scaled); pad heads 4..15 with zeros
        int p = t >> 2, h = t & 3;
        float acc = 0.f;
        for (int j = 0; j < Hdim; ++j) acc += tkh[h * 128 + j] * W_page[j * Pdim + p];
        ws[WS_A + p * 16 + h] = acc * SCALE;
        if (h == 0)
            for (int j2 = 4; j2 < 16; ++j2) ws[WS_A + p * 16 + j2] = 0.f;
    }
    if (t < 4) { // blend coeff + constant score term (scaled)
        float c = 0.f, k0 = 0.f;
        for (int j = 0; j < Hdim; ++j) {
            c  += tkh[t * 128 + j] * W_blend[j];
            k0 += tkh[t * 128 + j] * bsum[j];
        }
        for (int d = 0; d < 32; ++d) k0 += q[t * 32 + d] * ipb[Hdim + t * 32 + d];
        ws[WS_CS + t] = c * SCALE;
        ws[WS_K0 + t] = k0 * SCALE;
    }
    if (t < Hdim) { // bv_eff = Wv (b_page+b_blend) + bv ; wvb = Wv W_blend
        float a1 = ipb[2 * Hdim + t], a2 = 0.f;
        for (int j = 0; j < Hdim; ++j) {
            float w = ipw[(2 * Hdim + t) * Hdim + j];
            a1 += w * bsum[j];
            a2 += w * W_blend[j];
        }
        bveff[t] = a1;
        wvb[t]   = a2;
    }
    __syncthreads();

    if (t < Hdim) { // attended constant = Wout bv_eff + bo
        float acc = opb[t];
        for (int r = 0; r < Hdim; ++r) acc += opw[t * Hdim + r] * bveff[r];
        attc[t] = acc;
    }
    __syncthreads();

    if (t < 32) { // h1 constant
        float acc = b1[t];
        for (int i = 0; i < Hdim; ++i) acc += W1[t * Hdim + i] * attc[i];
        ws[WS_H1C + t] = acc;
    }

    // Y = W1 @ Wout  (32 x H)
    for (int e = t; e < 32 * Hdim; e += 256) {
        int o = e >> 7, hd = e & 127;
        float acc = 0.f;
        for (int i = 0; i < Hdim; ++i) acc += W1[o * Hdim + i] * opw[i * Hdim + hd];
        Y[e] = acc;
    }
    // Wvp = Wv @ W_page (H x P)
    for (int e = t; e < Hdim * Pdim; e += 256) {
        int r = e >> 6, p = e & 63;
        float acc = 0.f;
        for (int j = 0; j < Hdim; ++j) acc += ipw[(2 * Hdim + r) * Hdim + j] * W_page[j * Pdim + p];
        Wvp[e] = acc;
    }
    __syncthreads();

    // Gt[(h*64+p)*32 + o] = sum_d Y[o][h*32+d] * Wvp[h*32+d][p]
    for (int e = t; e < 8192; e += 256) {
        int o = e & 31, hp = e >> 5;
        int h = hp >> 6, p = hp & 63;
        float acc = 0.f;
        for (int d = 0; d < 32; ++d)
            acc += Y[o * 128 + h * 32 + d] * Wvp[(h * 32 + d) * 64 + p];
        ws[WS_GT + hp * 32 + o] = acc;
    }
    if (t < 128) { // U[o][h]
        int o = t >> 2, h = t & 3;
        float acc = 0.f;
        for (int d = 0; d < 32; ++d) acc += Y[o * 128 + h * 32 + d] * wvb[h * 32 + d];
        ws[WS_UPB + o * 4 + h] = acc;
    }
}

// ---------------------------------------------------------------------------
// Kernel 2: one workgroup per batch element. 256 threads = 8 wave32.
// LDS stride 68 floats: conflict-free for both WMMA fragment loads (row-major,
// lanes differ by row -> banks 4*row+col) and ps reduction (consecutive p).
// ---------------------------------------------------------------------------
#define SSTR 68

__global__ __launch_bounds__(256) void pch_main_kernel(
    const float* __restrict__ SP,   // (B,128,64)
    const float* __restrict__ BH,   // (B,128,1)
    const float* __restrict__ ws,
    const float* __restrict__ W2,   // (1,32)
    const float* __restrict__ b2,   // (1)
    float* __restrict__ out)        // (B,1)
{
    __shared__ float sS[Ndim * SSTR];   // 34.8 KB page tile
    __shared__ float sBlend[Ndim];
    __shared__ float sA[Pdim * 16];     // score weights, B-matrix layout
    __shared__ float sAttn[4 * Ndim];
    __shared__ float sPs[4 * Pdim];     // weighted page sums (flat = ps_flat)
    __shared__ float sPb[4];

    const int b = blockIdx.x;
    const int t = threadIdx.x;
    const float* Sg = SP + (size_t)b * (Ndim * Pdim);

    // ---- stage S_b (8192 floats) as float4, blend, and A into LDS ----
    for (int i = 0; i < 8; ++i) {
        int qi  = i * 256 + t;          // float4 index 0..2047
        int row = qi >> 4, c4 = qi & 15;
        float4 v = ((const float4*)Sg)[qi];
        *(float4*)&sS[row * SSTR + c4 * 4] = v;   // 272B row pitch -> 16B aligned
    }
    if (t < Ndim) sBlend[t] = BH[b * Ndim + t];
    for (int i = 0; i < 4; ++i) sA[i * 256 + t] = ws[WS_A + i * 256 + t];
    __syncthreads();

    // ---- scores = S(128x64) @ A(64x16) via 16x V_WMMA_F32_16X16X4_F32/wave ----
    const int w    = t >> 5;
    const int lane = t & 31;
    const int lo   = lane & 15;
    const bool hi  = lane >= 16;
    const int rowb = (w * 16 + lo) * SSTR;

    v8f acc = {};
    for (int kk = 0; kk < 16; ++kk) {
        int c0 = 4 * kk + (hi ? 2 : 0);
        v2f a, bb;
        a[0]  = sS[rowb + c0];
        a[1]  = sS[rowb + c0 + 1];
        bb[0] = sA[c0 * 16 + lo];
        bb[1] = sA[(c0 + 1) * 16 + lo];
        acc = __builtin_amdgcn_wmma_f32_16x16x4_f32(
            false, a, false, bb, (short)0, acc, false, false);
    }
    if (lo < 4) {  // columns 0..3 hold the 4 heads; add blend + const terms
        float cS = ws[WS_CS + lo], k0 = ws[WS_K0 + lo];
        for (int r = 0; r < 8; ++r) {
            int n = w * 16 + r + (hi ? 8 : 0);
            sAttn[lo * Ndim + n] = acc[r] + cS * sBlend[n] + k0;
        }
    }
    __syncthreads();

    // ---- softmax over N=128 per head (wave w handles head w) ----
    if (w < 4) {
        float v0[4], m = -3.4e38f;
        for (int i = 0; i < 4; ++i) {
            v0[i] = sAttn[w * Ndim + lane + 32 * i];
            m = fmaxf(m, v0[i]);
        }
        for (int off = 16; off; off >>= 1) m = fmaxf(m, __shfl_xor(m, off, 32));
        float s = 0.f;
        for (int i = 0; i < 4; ++i) { v0[i] = __expf(v0[i] - m); s += v0[i]; }
        for (int off = 16; off; off >>= 1) s += __shfl_xor(s, off, 32);
        float inv = 1.f / s, pb = 0.f;
        for (int i = 0; i < 4; ++i) {
            float a2 = v0[i] * inv;
            sAttn[w * Ndim + lane + 32 * i] = a2;
            pb += a2 * sBlend[lane + 32 * i];
        }
        for (int off = 16; off; off >>= 1) pb += __shfl_xor(pb, off, 32);
        if (lane == 0) sPb[w] = pb;
    }
    __syncthreads();

    // ---- ps[h][p] = sum_n attn[h][n] * S[n][p] ----
    {
        int h = t >> 6, p = t & 63;
        float a2 = 0.f;
        for (int n = 0; n < Ndim; ++n) a2 += sAttn[h * Ndim + n] * sS[n * SSTR + p];
        sPs[h * Pdim + p] = a2;
    }
    __syncthreads();

    // ---- h1 = relu(Gt^T ps + U pb + h1c); out = sigmoid(W2 h1 + b2) ----
    if (t < 32) {  // wave 0
        float a3 = ws[WS_H1C + t];
        for (int h = 0; h < 4; ++h) a3 += ws[WS_UPB + t * 4 + h] * sPb[h];
        for (int k = 0; k < 256; ++k) a3 += ws[WS_GT + k * 32 + t] * sPs[k];  // coalesced, L2-hot
        float contrib = W2[t] * fmaxf(a3, 0.f);
        for (int off = 16; off; off >>= 1) contrib += __shfl_xor(contrib, off, 32);
        if (t == 0) out[b] = 1.f / (1.f + __expf(-(contrib + b2[0])));
    }
}

extern "C" void kernel_launch(void* const* d_in, const int* in_sizes, int n_in,
                              void* d_out, int out_size, void* d_ws, size_t ws_size,
                              hipStream_t stream) {
    (void)in_sizes; (void)n_in; (void)out_size; (void)ws_size;
    const float* state_pages = (const float*)d_in[0];
    const float* blend_hist  = (const float*)d_in[1];
    const float* query       = (const float*)d_in[2];
    const float* W_page      = (const float*)d_in[3];
    const float* b_page      = (const float*)d_in[4];
    const float* W_blend     = (const float*)d_in[5];
    const float* b_blend     = (const float*)d_in[6];
    const float* in_proj_w   = (const float*)d_in[7];
    const float* in_proj_b   = (const float*)d_in[8];
    const float* out_proj_w  = (const float*)d_in[9];
    const float* out_proj_b  = (const float*)d_in[10];
    const float* W1          = (const float*)d_in[11];
    const float* b1          = (const float*)d_in[12];
    const float* W2          = (const float*)d_in[13];
    const float* b2          = (const float*)d_in[14];
    float* ws  = (float*)d_ws;
    float* out = (float*)d_out;

    pch_prep_kernel<<<1, 256, 0, stream>>>(query, W_page, b_page, W_blend, b_blend,
                                           in_proj_w, in_proj_b, out_proj_w, out_proj_b,
                                           W1, b1, ws);
    pch_main_kernel<<<4096, 256, 0, stream>>>(state_pages, blend_hist, ws, W2, b2, out);
}